// GHMC_Two_Stage_Loss_57157424775633
// MI455X (gfx1250) — compile-verified
//
#include <hip/hip_runtime.h>

#define GHM_BINS 10
#define GHM_C    64

typedef __attribute__((ext_vector_type(2))) float v2f;
typedef __attribute__((ext_vector_type(8))) float v8f;

// ---------------------------------------------------------------------------
// Kernel 0: zero the global accumulators in workspace (d_ws is poisoned).
// ---------------------------------------------------------------------------
__global__ void ghmc_init(unsigned* __restrict__ g_cnt, double* __restrict__ g_sum) {
    int t = threadIdx.x;
    if (t < GHM_BINS) {
        g_cnt[t] = 0u;
        g_sum[t] = 0.0;
    }
}

// ---------------------------------------------------------------------------
// Kernel 1: streaming pass over x. One lane per row (wave32 -> 32 rows/wave).
// Row is register-resident (16 x float4 = 64 floats); single read of HBM.
// Accumulate per-bin {count, ce_sum} in LDS, flush once per block to global.
// ---------------------------------------------------------------------------
__global__ __launch_bounds__(256) void ghmc_pass1(
    const float* __restrict__ x,
    const int*   __restrict__ target,
    const float* __restrict__ weight,
    const int*   __restrict__ stage_p,
    int N,
    unsigned* __restrict__ g_cnt,
    double*   __restrict__ g_sum)
{
    __shared__ float    ls_sum[GHM_BINS];
    __shared__ unsigned ls_cnt[GHM_BINS];
    if (threadIdx.x < GHM_BINS) {
        ls_sum[threadIdx.x] = 0.0f;
        ls_cnt[threadIdx.x] = 0u;
    }
    __syncthreads();

    const int stage = *stage_p;   // uniform scalar load
    const long long total  = (long long)N;
    const long long stride = (long long)gridDim.x * blockDim.x;

    for (long long r = (long long)blockIdx.x * blockDim.x + threadIdx.x;
         r < total; r += stride) {

        const float4* __restrict__ row = (const float4*)(x + r * (long long)GHM_C);

        float4 v[16];
        #pragma unroll
        for (int i = 0; i < 16; ++i) v[i] = row[i];   // 16x global_load_b128

        // row max (in-register tree)
        float mx = v[0].x, my = v[0].y, mz = v[0].z, mw = v[0].w;
        #pragma unroll
        for (int i = 1; i < 16; ++i) {
            mx = fmaxf(mx, v[i].x);
            my = fmaxf(my, v[i].y);
            mz = fmaxf(mz, v[i].z);
            mw = fmaxf(mw, v[i].w);
        }
        const float m = fmaxf(fmaxf(mx, my), fmaxf(mz, mw));

        // sum of exp(x - m)
        float sx = 0.f, sy = 0.f, sz = 0.f, sw = 0.f;
        #pragma unroll
        for (int i = 0; i < 16; ++i) {
            sx += __expf(v[i].x - m);
            sy += __expf(v[i].y - m);
            sz += __expf(v[i].z - m);
            sw += __expf(v[i].w - m);
        }
        const float s = (sx + sy) + (sz + sw);

        const int   t      = target[r];
        const float xt     = x[r * (long long)GHM_C + t];  // L0 hit: line just loaded
        const float d      = xt - m;
        const float log_pt = d - __logf(s);
        const float pt     = __expf(d) / s;

        const float ce = (stage == 1) ? (-log_pt) : (-weight[t] * log_pt);

        const float g = fabsf(pt - 1.0f);
        int b = (int)(g * 9.9999f);            // floor(g * (BINS - 1e-4))
        b = (b < 0) ? 0 : ((b > GHM_BINS - 1) ? GHM_BINS - 1 : b);

        atomicAdd(&ls_cnt[b], 1u);             // ds_add_u32
        atomicAdd(&ls_sum[b], ce);             // ds_add_f32
    }

    __syncthreads();
    if (threadIdx.x < GHM_BINS) {
        const unsigned c = ls_cnt[threadIdx.x];
        if (c) {
            atomicAdd(&g_cnt[threadIdx.x], c);
            atomicAdd(&g_sum[threadIdx.x], (double)ls_sum[threadIdx.x]); // global_atomic_add_f64
        }
    }
}

// ---------------------------------------------------------------------------
// Kernel 2: single wave32 epilogue.
//   nonempty = #bins with count>0 ; beta[b] = 1/max(count[b]*nonempty, 1e-4)
//   loss = (1/N) * sum_b beta[b] * ce_sum[b]
// The K=10 f32 dot product is computed with three V_WMMA_F32_16X16X4_F32 ops:
//   A row 0 = beta chunk (lane0 holds K=0,1 ; lane16 holds K=2,3)
//   B col 0 = ce_sum chunk (same K striping), all other entries zero.
//   D[0,0] lands in lane 0, element 0 of the f32 accumulator.
// EXEC is all-1s (32 threads, uniform control flow) as WMMA requires.
// ---------------------------------------------------------------------------
__global__ void ghmc_pass2(const unsigned* __restrict__ g_cnt,
                           const double*   __restrict__ g_sum,
                           float* __restrict__ out, int N)
{
    const int lane = threadIdx.x;   // 0..31

    // All addresses are uniform -> scalar loads; every lane holds all bins.
    float cnt[12], cs[12];
    #pragma unroll
    for (int b = 0; b < 12; ++b) {
        cnt[b] = (b < GHM_BINS) ? (float)g_cnt[b] : 0.0f;
        cs[b]  = (b < GHM_BINS) ? (float)g_sum[b] : 0.0f;
    }

    float nonempty = 0.0f;
    #pragma unroll
    for (int b = 0; b < GHM_BINS; ++b)
        nonempty += (cnt[b] > 0.0f) ? 1.0f : 0.0f;

    float beta[12];
    #pragma unroll
    for (int b = 0; b < 12; ++b) {
        const float gd = fmaxf(cnt[b] * nonempty, 1.0e-4f);  // MIN_GD
        beta[b] = (b < GHM_BINS) ? (1.0f / gd) : 0.0f;
    }

    v8f acc = {};
    #pragma unroll
    for (int c = 0; c < 3; ++c) {
        const float a0 = (lane == 0) ? beta[4*c + 0] : ((lane == 16) ? beta[4*c + 2] : 0.0f);
        const float a1 = (lane == 0) ? beta[4*c + 1] : ((lane == 16) ? beta[4*c + 3] : 0.0f);
        const float b0 = (lane == 0) ? cs[4*c + 0]   : ((lane == 16) ? cs[4*c + 2]   : 0.0f);
        const float b1 = (lane == 0) ? cs[4*c + 1]   : ((lane == 16) ? cs[4*c + 3]   : 0.0f);
        v2f A; A[0] = a0; A[1] = a1;
        v2f B; B[0] = b0; B[1] = b1;
        // (neg_a, A, neg_b, B, c_mod, C, reuse_a, reuse_b)
        acc = __builtin_amdgcn_wmma_f32_16x16x4_f32(false, A, false, B,
                                                    (short)0, acc, false, false);
    }

    if (lane == 0) out[0] = acc[0] / (float)N;
}

// ---------------------------------------------------------------------------
extern "C" void kernel_launch(void* const* d_in, const int* in_sizes, int n_in,
                              void* d_out, int out_size, void* d_ws, size_t ws_size,
                              hipStream_t stream) {
    const float* x      = (const float*)d_in[0];
    const int*   target = (const int*)  d_in[1];
    const float* weight = (const float*)d_in[2];
    const int*   stage  = (const int*)  d_in[3];
    const int N = in_sizes[1];          // target has N elements
    float* out = (float*)d_out;

    double*   g_sum = (double*)d_ws;                                  // 10 used (16 reserved)
    unsigned* g_cnt = (unsigned*)((char*)d_ws + 16 * sizeof(double)); // 10 used

    ghmc_init<<<1, 32, 0, stream>>>(g_cnt, g_sum);

    const int threads = 256;
    long long want = ((long long)N + threads - 1) / threads;
    int blocks = (want > 8192) ? 8192 : (want < 1 ? 1 : (int)want);
    ghmc_pass1<<<blocks, threads, 0, stream>>>(x, target, weight, stage, N, g_cnt, g_sum);

    ghmc_pass2<<<1, 32, 0, stream>>>(g_cnt, g_sum, out, N);
}